// DecoupSISO_BlockSSM_38053410242941
// MI455X (gfx1250) — compile-verified
//
#include <hip/hip_runtime.h>
#include <hip/hip_fp16.h>

typedef __attribute__((ext_vector_type(2))) float v2f;
typedef __attribute__((ext_vector_type(4))) float v4f;
typedef __attribute__((ext_vector_type(8))) float v8f;

#define NSTEPS 512
#define BATCH  512
#define NYC    32
#define HH     8
#define NXX    256
#define WAVES_PER_BLOCK 4
#define TOTAL_WAVES ((BATCH / 16) * NYC)   // 32 batch tiles x 32 channels = 1024

// Branch-free tanh-GELU: gelu(x) = x * sigmoid(2*c0*(x + c1*x^3))
// using raw v_exp_f32 / v_rcp_f32 (no ocml branchy tanhf).
__device__ __forceinline__ float gelu_fast(float x) {
  const float A = -2.0f * 0.7978845608028654f * 1.4426950408889634f; // -2*c0*log2(e)
  const float c1 = 0.044715f;
  float x2   = x * x;
  float poly = x * __builtin_fmaf(c1, x2, 1.0f);        // x + c1*x^3
  float e    = __builtin_amdgcn_exp2f(A * poly);        // exp(-2*c0*poly)
  return x * __builtin_amdgcn_rcpf(1.0f + e);           // x * sigmoid(...)
}

// Half-wave swap in the VALU: lane i <-> lane i^16 via v_permlanex16_b32 with
// identity selectors (no LDS pipe, no DScnt wait -- unlike ds_bpermute).
__device__ __forceinline__ float swap16(float x) {
  int xi = __float_as_int(x);
  int r  = __builtin_amdgcn_permlanex16(xi, xi, 0x76543210, 0xfedcba98,
                                        /*fi=*/false, /*bound_ctrl=*/false);
  return __int_as_float(r);
}

// One K=4 slice of D(16x16,f32) += A(16x4,f32) * B(4x16,f32)
__device__ __forceinline__ v8f wmma_f32k4(v2f a, v2f b, v8f c) {
  return __builtin_amdgcn_wmma_f32_16x16x4_f32(
      /*neg_a=*/false, a, /*neg_b=*/false, b,
      /*c_mod=*/(short)0, c, /*reuse_a=*/false, /*reuse_b=*/false);
}

// A operand: W^T in f32 A-layout. M(=lane&15) = output feature (rows 8..15 zero),
// K-block jbase: VGPR0 holds K = jbase+2*hi, VGPR1 holds K = jbase+2*hi+1.
// W is the 8x8 [in][out] matrix, so A[m][j] = W[j*8 + m].
__device__ __forceinline__ v2f make_A_block(const float* __restrict__ W,
                                            int m, int hi, int jbase) {
  v2f a = {};
  if (m < 8) {
    const int j0 = jbase + 2 * hi;
    a[0] = W[j0 * 8 + m];
    a[1] = W[(j0 + 1) * 8 + m];
  }
  return a;
}

// Repack D (feature r in VGPR r, batch in lanes 0-15; rows 8-15 are zero) into
// the two B operands of the next layer (B layout: lane = batch column, lo lanes
// hold K = {jb, jb+1}, hi lanes K = {jb+2, jb+3}). Features 2,3,6,7 move to the
// hi lane-half via v_permlanex16; GELU (optional) is applied after the repack so
// every lane does only 4 activations, all useful.
template <bool GELU>
__device__ __forceinline__ void d_to_b(v8f d, bool lo_half, v2f& b0, v2f& b1) {
  const float s2 = swap16(d[2]);
  const float s3 = swap16(d[3]);
  const float s6 = swap16(d[6]);
  const float s7 = swap16(d[7]);
  float e0 = lo_half ? d[0] : s2;
  float e1 = lo_half ? d[1] : s3;
  float e4 = lo_half ? d[4] : s6;
  float e5 = lo_half ? d[5] : s7;
  if (GELU) { e0 = gelu_fast(e0); e1 = gelu_fast(e1); e4 = gelu_fast(e4); e5 = gelu_fast(e5); }
  b0[0] = e0; b0[1] = e1;
  b1[0] = e4; b1[1] = e5;
}

__global__ __launch_bounds__(WAVES_PER_BLOCK * 32)
void decoup_siso_ssm_kernel(const float* __restrict__ x0,
                            const float* __restrict__ Uf,
                            const float* __restrict__ Wfx,
                            const float* __restrict__ Wfu0,
                            const float* __restrict__ Wfu1,
                            const float* __restrict__ Wfu2,
                            const float* __restrict__ Wfy,
                            float* __restrict__ outX,
                            float* __restrict__ outY,
                            float* __restrict__ outFU) {
  const int lane = threadIdx.x & 31;
  const int wave = blockIdx.x * WAVES_PER_BLOCK + (threadIdx.x >> 5);  // 0..1023

  const int b0 = (wave & 31) * 16;    // batch tile base (32 tiles of 16)
  const int k  = wave >> 5;           // channel (wave-uniform)

  const int m    = lane & 15;         // A-row (output feature) / B-column (batch)
  const int hi   = lane >> 4;
  const bool lo_half = (lane < 16);

  // ---- weights, transposed into f32 A layout; two K=4 blocks per layer ----
  const float* Wx0 = Wfx + (size_t)(k * 3 + 0) * 64;
  const float* Wx1 = Wfx + (size_t)(k * 3 + 1) * 64;
  const float* Wx2 = Wfx + (size_t)(k * 3 + 2) * 64;
  const float* Wu1 = Wfu1 + (size_t)k * 64;
  const float* Wu2 = Wfu2 + (size_t)k * 64;
  const v2f Afx0b0 = make_A_block(Wx0, m, hi, 0), Afx0b1 = make_A_block(Wx0, m, hi, 4);
  const v2f Afx1b0 = make_A_block(Wx1, m, hi, 0), Afx1b1 = make_A_block(Wx1, m, hi, 4);
  const v2f Afx2b0 = make_A_block(Wx2, m, hi, 0), Afx2b1 = make_A_block(Wx2, m, hi, 4);
  const v2f Afu1b0 = make_A_block(Wu1, m, hi, 0), Afu1b1 = make_A_block(Wu1, m, hi, 4);
  const v2f Afu2b0 = make_A_block(Wu2, m, hi, 0), Afu2b1 = make_A_block(Wu2, m, hi, 4);

  float wfu0r[8];                      // wave-uniform -> scalarizable
  #pragma unroll
  for (int j = 0; j < 8; ++j) wfu0r[j] = Wfu0[k * 8 + j];
  float wfyr[8];
  #pragma unroll
  for (int j = 0; j < 8; ++j) wfyr[j] = Wfy[k * 8 + j];

  const int rowA = b0 + m;             // batch row = this lane's B column

  // ---- initial state straight into B layout (all lanes carry real data) ----
  v2f sB0, sB1;
  {
    const float* xp = x0 + (size_t)rowA * NXX + k * 8;
    sB0[0] = xp[0 + 2 * hi]; sB0[1] = xp[1 + 2 * hi];
    sB1[0] = xp[4 + 2 * hi]; sB1[1] = xp[5 + 2 * hi];
  }

  const v8f zero8 = {};

  for (int t = 0; t < NSTEPS; ++t) {
    // ---- fx(x): 3 linear layers, GELU on the two hidden ones ----
    v8f t1 = wmma_f32k4(Afx0b0, sB0, zero8);
    t1     = wmma_f32k4(Afx0b1, sB1, t1);
    v2f g1b0, g1b1; d_to_b<true>(t1, lo_half, g1b0, g1b1);
    v8f t2 = wmma_f32k4(Afx1b0, g1b0, zero8);
    t2     = wmma_f32k4(Afx1b1, g1b1, t2);
    v2f g2b0, g2b1; d_to_b<true>(t2, lo_half, g2b0, g2b1);
    v8f fxD = wmma_f32k4(Afx2b0, g2b0, zero8);
    fxD     = wmma_f32k4(Afx2b1, g2b1, fxD);

    // ---- fu(u): layer0 (scalar*vec + GELU) built directly in B layout ----
    const float u = Uf[(size_t)t * (BATCH * NYC) + (size_t)rowA * NYC + k];
    v2f z0b0, z0b1;
    z0b0[0] = gelu_fast(u * wfu0r[0 + 2 * hi]);
    z0b0[1] = gelu_fast(u * wfu0r[1 + 2 * hi]);
    z0b1[0] = gelu_fast(u * wfu0r[4 + 2 * hi]);
    z0b1[1] = gelu_fast(u * wfu0r[5 + 2 * hi]);
    v8f z1 = wmma_f32k4(Afu1b0, z0b0, zero8);
    z1     = wmma_f32k4(Afu1b1, z0b1, z1);
    v2f gzb0, gzb1; d_to_b<true>(z1, lo_half, gzb0, gzb1);
    // layer2 with C = fx output: xn = fx(x) + fu(u)
    v8f xnD = wmma_f32k4(Afu2b0, gzb0, fxD);
    xnD     = wmma_f32k4(Afu2b1, gzb1, xnD);

    // ---- outputs: this lane (lanes 0-15) owns batch row rowA, features 0-7
    //      contiguous in VGPRs -> two 128-bit stores per tensor ----
    if (lo_half) {
      float* px = outX  + (size_t)t * (BATCH * NXX) + (size_t)rowA * NXX + k * 8;
      float* pf = outFU + (size_t)t * (BATCH * NXX) + (size_t)rowA * NXX + k * 8;
      v4f xlo = {xnD[0], xnD[1], xnD[2], xnD[3]};
      v4f xhi = {xnD[4], xnD[5], xnD[6], xnD[7]};
      v4f flo = {xnD[0] - fxD[0], xnD[1] - fxD[1], xnD[2] - fxD[2], xnD[3] - fxD[3]};
      v4f fhi = {xnD[4] - fxD[4], xnD[5] - fxD[5], xnD[6] - fxD[6], xnD[7] - fxD[7]};
      *(v4f*)px       = xlo;
      *(v4f*)(px + 4) = xhi;
      *(v4f*)pf       = flo;
      *(v4f*)(pf + 4) = fhi;
      // y = <xn, Wfy>: per-lane dot product, no shuffles
      float y = 0.0f;
      #pragma unroll
      for (int r = 0; r < 8; ++r) y = __builtin_fmaf(xnD[r], wfyr[r], y);
      outY[(size_t)t * (BATCH * NYC) + (size_t)rowA * NYC + k] = y;
    }

    // prefetch next step's control input
    if (t + 1 < NSTEPS)
      __builtin_prefetch(Uf + (size_t)(t + 1) * (BATCH * NYC) + (size_t)rowA * NYC + k, 0, 1);

    // ---- carry state: xn (D layout) -> B layout for next step (no GELU) ----
    d_to_b<false>(xnD, lo_half, sB0, sB1);
  }
}

extern "C" void kernel_launch(void* const* d_in, const int* in_sizes, int n_in,
                              void* d_out, int out_size, void* d_ws, size_t ws_size,
                              hipStream_t stream) {
  (void)in_sizes; (void)n_in; (void)out_size; (void)d_ws; (void)ws_size;
  const float* x0   = (const float*)d_in[0];
  // d_in[1] = Yf is unused by the reference computation
  const float* Uf   = (const float*)d_in[2];
  const float* Wfx  = (const float*)d_in[3];
  const float* Wfu0 = (const float*)d_in[4];
  const float* Wfu1 = (const float*)d_in[5];
  const float* Wfu2 = (const float*)d_in[6];
  const float* Wfy  = (const float*)d_in[7];

  float* outX  = (float*)d_out;                                   // [512,512,256]
  float* outY  = outX + (size_t)NSTEPS * BATCH * NXX;             // [512,512,32]
  float* outFU = outY + (size_t)NSTEPS * BATCH * NYC;             // [512,512,256]

  const dim3 grid(TOTAL_WAVES / WAVES_PER_BLOCK);                 // 256 blocks
  const dim3 block(WAVES_PER_BLOCK * 32);                         // 128 threads

  decoup_siso_ssm_kernel<<<grid, block, 0, stream>>>(
      x0, Uf, Wfx, Wfu0, Wfu1, Wfu2, Wfy, outX, outY, outFU);
}